// SGCLayer_1692217115479
// MI455X (gfx1250) — compile-verified
//
#include <hip/hip_runtime.h>

// SGC layer: h = x @ W^T + b, then 3 hops of weighted SpMM (scatter-add).
// GEMM uses V_WMMA_F32_16X16X4_F32 (exact fp32). SpMM is L2-resident on
// MI455X (h=25.6MB + edges=32MB << 192MB L2) and uses native f32 atomics.

typedef __attribute__((ext_vector_type(2))) float v2f;
typedef __attribute__((ext_vector_type(8))) float v8f;

#define IN_F 128
#define OUT_F 64

// ---------------------------------------------------------------------------
// Kernel 1: h[N,64] = x[N,128] @ W^T + bias, one 16x16 tile per wave32.
// 32 chained V_WMMA_F32_16X16X4_F32 per tile (K=128, 4 per step).
// A row index is CLAMPED (not predicated) so the k-loop has zero exec-mask
// manipulation: each step is global_load_b64 + ds_load_b64 + v_wmma.
// ---------------------------------------------------------------------------
__global__ void __launch_bounds__(128)
sgc_gemm_wmma(const float* __restrict__ x,
              const float* __restrict__ W,    // [OUT_F, IN_F] row-major
              const float* __restrict__ bias, // [OUT_F]
              float* __restrict__ h,          // [N, OUT_F]
              int nNodes, int nMTiles)
{
    __shared__ float sW[OUT_F * IN_F]; // 32 KB of the 320 KB/WGP LDS

    for (int i = threadIdx.x; i < OUT_F * IN_F; i += blockDim.x)
        sW[i] = W[i];
    __syncthreads();

    const int wave = threadIdx.x >> 5;
    const int lane = threadIdx.x & 31;
    const int tile = blockIdx.x * 4 + wave;       // 4 waves per block
    const int mtile = tile >> 2;                  // OUT_F/16 == 4 n-tiles
    const int ntile = tile & 3;
    if (mtile >= nMTiles) return;                 // wave-uniform guard

    const int col = ntile * 16 + (lane & 15);     // C/D: N = lane%16
    const int kShift = (lane < 16) ? 0 : 2;       // A/B: K pair {0,1} vs {2,3}

    // A: M = lane%16 (both lane halves). Clamp to a valid row: garbage from
    // clamped rows only affects C rows that the store guard drops.
    int arow = mtile * 16 + (lane & 15);
    arow = arow < nNodes ? arow : (nNodes - 1);

    // C accumulator seeded with bias (bias depends only on column).
    v8f c;
    const float b0 = bias[col];
#pragma unroll
    for (int r = 0; r < 8; ++r) c[r] = b0;

    const float* aPtr = x + (size_t)arow * IN_F + kShift;  // 8B aligned
    const float* bPtr = sW + col * IN_F + kShift;          // 8B aligned

#pragma unroll
    for (int k = 0; k < IN_F; k += 4) {
        const v2f a = *(const v2f*)(aPtr + k);  // A[row][k+kShift, +1]
        const v2f b = *(const v2f*)(bPtr + k);  // W[col][k+kShift, +1]
        // D = A x B + C, fp32 exact
        c = __builtin_amdgcn_wmma_f32_16x16x4_f32(
                /*neg_a=*/false, a, /*neg_b=*/false, b,
                /*c_mod=*/(short)0, c, /*reuse_a=*/false, /*reuse_b=*/false);
    }

    // C/D layout: VGPR r -> row r (lanes 0-15) or r+8 (lanes 16-31).
    const int rowBase = mtile * 16 + ((lane >= 16) ? 8 : 0);
#pragma unroll
    for (int r = 0; r < 8; ++r) {
        const int orow = rowBase + r;
        if (orow < nNodes)
            h[(size_t)orow * OUT_F + col] = c[r];
    }
}

// ---------------------------------------------------------------------------
// Kernel 2: out[dst] += w[e] * hin[src[e]]. 16 lanes per edge, 4 feats/lane:
// one coalesced float4 gather (16 lanes cover the full 256B row) and 4
// native global_atomic_add_f32 to L2.
// ---------------------------------------------------------------------------
__global__ void __launch_bounds__(256)
sgc_spmm(const long long* __restrict__ src,
         const long long* __restrict__ dst,
         const float* __restrict__ ew,
         const float* __restrict__ hin,
         float* __restrict__ hout,
         int nEdges)
{
    const long long g = (long long)blockIdx.x * blockDim.x + threadIdx.x;
    const int e = (int)(g >> 4);
    if (e >= nEdges) return;
    const int q = (int)(g & 15);                  // feature quad 0..15

    const int  s  = (int)src[e];
    const int  d  = (int)dst[e];
    const float we = ew[e];

    const float4 hv = *(const float4*)(hin + (size_t)s * OUT_F + q * 4);
    float* o = hout + (size_t)d * OUT_F + q * 4;
    unsafeAtomicAdd(o + 0, hv.x * we);
    unsafeAtomicAdd(o + 1, hv.y * we);
    unsafeAtomicAdd(o + 2, hv.z * we);
    unsafeAtomicAdd(o + 3, hv.w * we);
}

// ---------------------------------------------------------------------------
extern "C" void kernel_launch(void* const* d_in, const int* in_sizes, int n_in,
                              void* d_out, int out_size, void* d_ws, size_t ws_size,
                              hipStream_t stream)
{
    const float*     x  = (const float*)d_in[0];     // [N, 128]
    const long long* ei = (const long long*)d_in[1]; // [2, E] int64
    const float*     ew = (const float*)d_in[2];     // [E]
    const float*     Ww = (const float*)d_in[3];     // [64, 128]
    const float*     Wb = (const float*)d_in[4];     // [64]
    float* out = (float*)d_out;

    const int nNodes = in_sizes[0] / IN_F;
    const int nEdges = in_sizes[2];
    const long long* src = ei;
    const long long* dst = ei + nEdges;

    float* h0 = (float*)d_ws;                        // nNodes*64 fp32 (25.6 MB)
    const size_t hBytes = (size_t)nNodes * OUT_F * sizeof(float);

    // --- projection: x @ W^T + b -> h0 (WMMA) ---
    const int nMTiles = (nNodes + 15) / 16;
    const int tiles   = nMTiles * (OUT_F / 16);
    sgc_gemm_wmma<<<(tiles + 3) / 4, 128, 0, stream>>>(x, Ww, Wb, h0, nNodes, nMTiles);

    // --- 3 hops of SpMM, ping-pong h0 <-> out ---
    const long long tot = (long long)nEdges * 16;
    const int eBlocks = (int)((tot + 255) / 256);

    hipMemsetAsync(out, 0, hBytes, stream);
    sgc_spmm<<<eBlocks, 256, 0, stream>>>(src, dst, ew, h0, out, nEdges);

    hipMemsetAsync(h0, 0, hBytes, stream);
    sgc_spmm<<<eBlocks, 256, 0, stream>>>(src, dst, ew, out, h0, nEdges);

    hipMemsetAsync(out, 0, hBytes, stream);
    sgc_spmm<<<eBlocks, 256, 0, stream>>>(src, dst, ew, h0, out, nEdges);
}